// GaussianTriPlane_4226247819637
// MI455X (gfx1250) — compile-verified
//
#include <hip/hip_runtime.h>

// ---------------------------------------------------------------------------
// GaussianTriPlane for MI455X (gfx1250, wave32)
//   scatter-splat (hw fp32 atomics) -> layernorm (2-stage reduce) ->
//   separable 5x5 gaussian blur done as chained V_WMMA_F32_16X16X4_F32
//   with banded-kernel weights from a compile-time __constant__ table.
// ---------------------------------------------------------------------------

#define RES   512
#define HW    (RES * RES)          // 262144
#define NCH   32
#define CHW   (NCH * HW)           // 8388608
#define NG    500000
#define RBLK  1024                 // reduction partial blocks

typedef __attribute__((ext_vector_type(2))) float v2f;
typedef __attribute__((ext_vector_type(8))) float v8f;

// normalized 1D gaussian kernel, sigma=1: exp(-0.5*{4,1,0,1,4})/sum
constexpr float gwc(int d) {
  return (d == 0) ? 0.054488685f
       : (d == 1) ? 0.244201342f
       : (d == 2) ? 0.402619950f
       : (d == 3) ? 0.244201342f
       : (d == 4) ? 0.054488685f
       : 0.0f;                       // 0 outside band
}

// Banded matrix Band[k,n] = w[k-n], laid out per (kk-chunk, lane) exactly as
// the 32-bit WMMA A/B operand layout needs it:
//   lane L -> n = L%16, K pair = kk*4 + (L/16)*2 + {0,1}
// Same table serves the h-pass B matrix and the v-pass A matrix.
struct WTab { float v[5][32][2]; };
constexpr WTab makeWTab() {
  WTab t{};
  for (int kk = 0; kk < 5; ++kk)
    for (int l = 0; l < 32; ++l)
      for (int j = 0; j < 2; ++j) {
        int kb = kk * 4 + (l >> 4) * 2 + j;
        int n  = l & 15;
        t.v[kk][l][j] = gwc(kb - n);
      }
  return t;
}
__device__ __constant__ WTab WBAND = makeWTab();

__global__ void k_zero(float* __restrict__ p, int n) {
  int i = blockIdx.x * blockDim.x + threadIdx.x;
  int stride = gridDim.x * blockDim.x;
  for (; i < n; i += stride) p[i] = 0.0f;
}

// one wave per gaussian, lane = channel; 4 bilinear-corner atomics per lane
__global__ void __launch_bounds__(256)
k_scatter(const float* __restrict__ xyz, const float* __restrict__ feats,
          float* __restrict__ acc, float* __restrict__ cnt, int plane) {
  long long tid = (long long)blockIdx.x * blockDim.x + threadIdx.x;
  int n = (int)(tid >> 5);
  int c = (int)(tid & 31);
  if (n >= NG) return;

  int ui = (plane == 2) ? 1 : 0;    // xy:(x,y)  xz:(x,z)  yz:(y,z)
  int vi = (plane == 0) ? 1 : 2;
  float u = xyz[(long long)n * 3 + ui];
  float v = xyz[(long long)n * 3 + vi];
  u = (u + 1.0f) * 0.5f;
  v = (v + 1.0f) * 0.5f;
  u = fminf(fmaxf(u, 0.0f), 0.999f) * (float)(RES - 1);
  v = fminf(fmaxf(v, 0.0f), 0.999f) * (float)(RES - 1);

  float fu = floorf(u), fv = floorf(v);
  int x0 = min(max((int)fu, 0), RES - 1);
  int x1 = min(max((int)fu + 1, 0), RES - 1);
  int y0 = min(max((int)fv, 0), RES - 1);
  int y1 = min(max((int)fv + 1, 0), RES - 1);
  float wx0 = fminf(fmaxf((float)x1 - u, 0.0f), 1.0f);
  float wx1 = fminf(fmaxf(u - (float)x0, 0.0f), 1.0f);
  float wy0 = fminf(fmaxf((float)y1 - v, 0.0f), 1.0f);
  float wy1 = fminf(fmaxf(v - (float)y0, 0.0f), 1.0f);

  float f = feats[(long long)n * (3 * NCH) + plane * NCH + c];
  float* __restrict__ ac = acc + (size_t)c * HW;
  unsafeAtomicAdd(&ac[y0 * RES + x0], f * wx0 * wy0);
  unsafeAtomicAdd(&ac[y1 * RES + x0], f * wx0 * wy1);
  unsafeAtomicAdd(&ac[y0 * RES + x1], f * wx1 * wy0);
  unsafeAtomicAdd(&ac[y1 * RES + x1], f * wx1 * wy1);
  if (c == 0) {                      // cnt gets +1 per (gaussian, corner)
    unsafeAtomicAdd(&cnt[y0 * RES + x0], 1.0f);
    unsafeAtomicAdd(&cnt[y1 * RES + x0], 1.0f);
    unsafeAtomicAdd(&cnt[y0 * RES + x1], 1.0f);
    unsafeAtomicAdd(&cnt[y1 * RES + x1], 1.0f);
  }
}

__global__ void __launch_bounds__(256)
k_reduce1(const float* __restrict__ acc, const float* __restrict__ cnt,
          double* __restrict__ part) {
  __shared__ double ss[256], sq[256];
  double s = 0.0, q = 0.0;
  int stride = gridDim.x * blockDim.x;
  for (int i = blockIdx.x * blockDim.x + threadIdx.x; i < CHW; i += stride) {
    float val = acc[i] / (cnt[i & (HW - 1)] + 1e-6f);
    s += (double)val;
    q += (double)val * (double)val;
  }
  ss[threadIdx.x] = s; sq[threadIdx.x] = q;
  __syncthreads();
  for (int o = 128; o > 0; o >>= 1) {
    if ((int)threadIdx.x < o) {
      ss[threadIdx.x] += ss[threadIdx.x + o];
      sq[threadIdx.x] += sq[threadIdx.x + o];
    }
    __syncthreads();
  }
  if (threadIdx.x == 0) {
    part[2 * blockIdx.x + 0] = ss[0];
    part[2 * blockIdx.x + 1] = sq[0];
  }
}

__global__ void __launch_bounds__(256)
k_reduce2(const double* __restrict__ part, int nblocks, float* __restrict__ stats) {
  __shared__ double ss[256], sq[256];
  double s = 0.0, q = 0.0;
  for (int i = threadIdx.x; i < nblocks; i += blockDim.x) {
    s += part[2 * i + 0];
    q += part[2 * i + 1];
  }
  ss[threadIdx.x] = s; sq[threadIdx.x] = q;
  __syncthreads();
  for (int o = 128; o > 0; o >>= 1) {
    if ((int)threadIdx.x < o) {
      ss[threadIdx.x] += ss[threadIdx.x + o];
      sq[threadIdx.x] += sq[threadIdx.x + o];
    }
    __syncthreads();
  }
  if (threadIdx.x == 0) {
    double mu  = ss[0] / (double)CHW;
    double var = sq[0] / (double)CHW - mu * mu;
    var = var < 0.0 ? 0.0 : var;
    stats[0] = (float)mu;
    stats[1] = (float)(1.0 / sqrt(var + 1e-5));
  }
}

__global__ void __launch_bounds__(256)
k_norm(const float* __restrict__ acc, const float* __restrict__ cnt,
       const float* __restrict__ lnw, const float* __restrict__ lnb,
       const float* __restrict__ stats, float* __restrict__ out) {
  float mu = stats[0], rs = stats[1];
  int stride = gridDim.x * blockDim.x;
  for (int i = blockIdx.x * blockDim.x + threadIdx.x; i < CHW; i += stride) {
    float val = acc[i] / (cnt[i & (HW - 1)] + 1e-6f);
    out[i] = (val - mu) * rs * lnw[i] + lnb[i];
  }
}

// Horizontal blur: D[16x16] = In[16x20] * Band[20x16], 5 chained K=4 f32 WMMAs.
// One wave per (channel, 16x16 tile). 4096 blocks * 8 waves == 32*32*32 tiles.
__global__ void __launch_bounds__(256)
k_hblur(const float* __restrict__ in, float* __restrict__ out) {
  int gtid = blockIdx.x * blockDim.x + threadIdx.x;
  int wave = gtid >> 5;
  int lane = threadIdx.x & 31;
  int tx = wave & 31, ty = (wave >> 5) & 31, cc = wave >> 10;
  int x0 = tx << 4, y0 = ty << 4;
  int m  = lane & 15;                 // A row / B-D column
  int hi = lane >> 4;                 // lane half selects K pair
  const float* __restrict__ pin = in + (size_t)cc * HW + (size_t)(y0 + m) * RES;
  v8f c8 = {};
#pragma unroll
  for (int kk = 0; kk < 5; ++kk) {
    int kb  = kk * 4 + hi * 2;        // this lane's two K indices: kb, kb+1
    int col = x0 - 2 + kb;
    v2f a;
    a.x = (col     >= 0 && col     < RES) ? pin[col]     : 0.0f;  // zero-pad
    a.y = (col + 1 >= 0 && col + 1 < RES) ? pin[col + 1] : 0.0f;
    v2f b = *(const v2f*)&WBAND.v[kk][lane][0];   // Band[k,n] = w[k-n]
    c8 = __builtin_amdgcn_wmma_f32_16x16x4_f32(
        false, a, false, b, (short)0, c8, false, false);
  }
  float* __restrict__ pout = out + (size_t)cc * HW;
#pragma unroll
  for (int v = 0; v < 8; ++v) {
    int r = y0 + v + hi * 8;          // D layout: VGPR v -> M = v + 8*hi
    pout[(size_t)r * RES + x0 + m] = c8[v];
  }
}

// Vertical blur + residual: D[16x16] = Band^T[16x20] * In[20x16], + plane.
__global__ void __launch_bounds__(256)
k_vblur(const float* __restrict__ in, const float* __restrict__ plane,
        float* __restrict__ out) {
  int gtid = blockIdx.x * blockDim.x + threadIdx.x;
  int wave = gtid >> 5;
  int lane = threadIdx.x & 31;
  int tx = wave & 31, ty = (wave >> 5) & 31, cc = wave >> 10;
  int x0 = tx << 4, y0 = ty << 4;
  int m  = lane & 15;
  int hi = lane >> 4;
  const float* __restrict__ pin = in + (size_t)cc * HW;
  int col = x0 + m;
  v8f c8 = {};
#pragma unroll
  for (int kk = 0; kk < 5; ++kk) {
    int kb = kk * 4 + hi * 2;
    v2f a = *(const v2f*)&WBAND.v[kk][lane][0];   // Band^T[m,k] = w[k-m]
    v2f b;
    int r0 = y0 - 2 + kb;
    b.x = (r0     >= 0 && r0     < RES) ? pin[(size_t)r0 * RES + col]       : 0.0f;
    b.y = (r0 + 1 >= 0 && r0 + 1 < RES) ? pin[(size_t)(r0 + 1) * RES + col] : 0.0f;
    c8 = __builtin_amdgcn_wmma_f32_16x16x4_f32(
        false, a, false, b, (short)0, c8, false, false);
  }
  const float* __restrict__ pp = plane + (size_t)cc * HW;
  float* __restrict__ po = out + (size_t)cc * HW;
#pragma unroll
  for (int v = 0; v < 8; ++v) {
    int r = y0 + v + hi * 8;
    size_t idx = (size_t)r * RES + x0 + m;
    po[idx] = c8[v] + pp[idx];
  }
}

extern "C" void kernel_launch(void* const* d_in, const int* in_sizes, int n_in,
                              void* d_out, int out_size, void* d_ws, size_t ws_size,
                              hipStream_t stream) {
  (void)in_sizes; (void)n_in; (void)out_size; (void)ws_size;
  const float* feats = (const float*)d_in[0];   // [N, 96]
  const float* xyz   = (const float*)d_in[1];   // [N, 3]
  const float* planes[3] = {(const float*)d_in[2], (const float*)d_in[3],
                            (const float*)d_in[4]};
  const float* lnw = (const float*)d_in[5];     // [32,512,512]
  const float* lnb = (const float*)d_in[6];
  float* out = (float*)d_out;                   // 3 * CHW

  char* ws = (char*)d_ws;                       // scratch layout (~68.2 MB)
  float*  acc   = (float*)(ws);                                   // CHW f32
  float*  nrm   = (float*)(ws + (size_t)CHW * 4);                 // CHW f32
  float*  cnt   = (float*)(ws + (size_t)CHW * 8);                 // HW  f32
  double* part  = (double*)(ws + (size_t)CHW * 8 + (size_t)HW * 4); // RBLK*2 f64
  float*  stats = (float*)((char*)part + (size_t)RBLK * 2 * sizeof(double));

  for (int p = 0; p < 3; ++p) {
    k_zero   <<<8192, 256, 0, stream>>>(acc, CHW);
    k_zero   <<<1024, 256, 0, stream>>>(cnt, HW);
    k_scatter<<<(NG * 32) / 256, 256, 0, stream>>>(xyz, feats, acc, cnt, p);
    k_reduce1<<<RBLK, 256, 0, stream>>>(acc, cnt, part);
    k_reduce2<<<1, 256, 0, stream>>>(part, RBLK, stats);
    k_norm   <<<8192, 256, 0, stream>>>(acc, cnt, lnw, lnb, stats, nrm);
    k_hblur  <<<4096, 256, 0, stream>>>(nrm, acc);          // reuse acc as temp
    k_vblur  <<<4096, 256, 0, stream>>>(acc, planes[p], out + (size_t)p * CHW);
  }
}